// GCN_60988535603684
// MI455X (gfx1250) — compile-verified
//
#include <hip/hip_runtime.h>

typedef __attribute__((ext_vector_type(2))) float v2f;
typedef __attribute__((ext_vector_type(8))) float v8f;

#define F 128  // IN_FEATS == HIDDEN == 128

// ---------------- degree / normalization ----------------
__global__ void deg_init_kernel(float* deg, int n) {
  int i = blockIdx.x * blockDim.x + threadIdx.x;
  if (i < n) deg[i] = 1.0f;  // self-loop contributes weight 1 to every node
}

__global__ void deg_edge_kernel(float* deg, const int* __restrict__ col,
                                const float* __restrict__ ew, int E) {
  int e = blockIdx.x * blockDim.x + threadIdx.x;
  if (e < E) atomicAdd(&deg[col[e]], ew[e]);
}

__global__ void dinv_kernel(float* deg, int n) {
  int i = blockIdx.x * blockDim.x + threadIdx.x;
  if (i < n) {
    float d = deg[i];
    deg[i] = (d > 0.0f) ? rsqrtf(d) : 0.0f;
  }
}

// ---------------- dense GEMM: OUT[n][j] = sum_k X[n][k] * W[j][k] ----------
// One wave per 16-row strip; computes all 8 column tiles (128 = 8*16).
// f32 WMMA 16x16x4: A lane L -> M=L&15, K=2*(L>>4)+{0,1}; B mirrors with N.
__global__ void __launch_bounds__(32)
gemm_wmma_f32(const float* __restrict__ X, const float* __restrict__ W,
              float* __restrict__ OUT, int nrows) {
  int nb = blockIdx.x * 16;
  if (nb >= nrows) return;                 // wave-uniform guard
  int lane = threadIdx.x & 31;
  int m    = lane & 15;
  int koff = (lane >> 4) << 1;             // 0 or 2
  int mrow = nb + m; if (mrow >= nrows) mrow = nrows - 1;
  const float* xrow = X + (size_t)mrow * F;

  v8f acc[8];
  v8f zero = {};
#pragma unroll
  for (int j = 0; j < 8; ++j) acc[j] = zero;

  for (int kb = 0; kb < F; kb += 4) {
    v2f a = *(const v2f*)(xrow + kb + koff);
#pragma unroll
    for (int j = 0; j < 8; ++j) {
      v2f b = *(const v2f*)(W + (size_t)(j * 16 + m) * F + kb + koff);
      acc[j] = __builtin_amdgcn_wmma_f32_16x16x4_f32(
          /*neg_a=*/false, a, /*neg_b=*/false, b,
          /*c_mod=*/(short)0, acc[j], /*reuse_a=*/false, /*reuse_b=*/false);
    }
  }

  int rbase = nb + ((lane >> 4) << 3);     // D: VGPR r -> row rbase+r, col m
#pragma unroll
  for (int j = 0; j < 8; ++j) {
#pragma unroll
    for (int r = 0; r < 8; ++r) {
      int rr = rbase + r;
      if (rr < nrows) OUT[(size_t)rr * F + j * 16 + m] = acc[j][r];
    }
  }
}

// ---------------- self-loop init: out[n] = dinv[n]^2 * h[n] ----------------
__global__ void selfloop_init_kernel(const float* __restrict__ h,
                                     const float* __restrict__ dinv,
                                     float* __restrict__ out, int n) {
  int gid = blockIdx.x * blockDim.x + threadIdx.x;
  int node = gid >> 5;
  if (node >= n) return;
  int lane = gid & 31;
  float d = dinv[node];
  float s = d * d;
  float4 v = ((const float4*)(h + (size_t)node * F))[lane];
  float4 o; o.x = s*v.x; o.y = s*v.y; o.z = s*v.z; o.w = s*v.w;
  ((float4*)(out + (size_t)node * F))[lane] = o;
}

// ---------------- edge scatter: out[col] += dinv[r]*ew*dinv[c] * h[row] ----
__global__ void agg_edge_kernel(const float* __restrict__ h,
                                const int* __restrict__ row,
                                const int* __restrict__ col,
                                const float* __restrict__ ew,
                                const float* __restrict__ dinv,
                                float* __restrict__ out, int E) {
  int gid = blockIdx.x * blockDim.x + threadIdx.x;
  int e = gid >> 5;
  if (e >= E) return;
  int lane = gid & 31;
  int r = row[e], c = col[e];
  float norm = dinv[r] * ew[e] * dinv[c];
  float4 v = ((const float4*)(h + (size_t)r * F))[lane];
  float* o = out + (size_t)c * F + lane * 4;
  atomicAdd(o + 0, norm * v.x);
  atomicAdd(o + 1, norm * v.y);
  atomicAdd(o + 2, norm * v.z);
  atomicAdd(o + 3, norm * v.w);
}

// ---------------- bias + relu (in place) -----------------------------------
__global__ void bias_relu_kernel(float* __restrict__ h, const float* __restrict__ b, int n) {
  int gid = blockIdx.x * blockDim.x + threadIdx.x;
  int node = gid >> 5;
  if (node >= n) return;
  int lane = gid & 31;
  float4 bv = ((const float4*)b)[lane];
  float4* p = (float4*)(h + (size_t)node * F) + lane;
  float4 v = *p;
  v.x = fmaxf(v.x + bv.x, 0.0f);
  v.y = fmaxf(v.y + bv.y, 0.0f);
  v.z = fmaxf(v.z + bv.z, 0.0f);
  v.w = fmaxf(v.w + bv.w, 0.0f);
  *p = v;
}

// ---------------- pooling ---------------------------------------------------
__global__ void zero_kernel(float* p, int n) {
  int i = blockIdx.x * blockDim.x + threadIdx.x;
  if (i < n) p[i] = 0.0f;
}

__global__ void pool_add_kernel(const float* __restrict__ h, const int* __restrict__ batch,
                                float* __restrict__ pool, float* __restrict__ cnt, int n) {
  int gid = blockIdx.x * blockDim.x + threadIdx.x;
  int node = gid >> 5;
  if (node >= n) return;
  int lane = gid & 31;
  int g = batch[node];
  float4 v = ((const float4*)(h + (size_t)node * F))[lane];
  float* p = pool + (size_t)g * F + lane * 4;
  atomicAdd(p + 0, v.x);
  atomicAdd(p + 1, v.y);
  atomicAdd(p + 2, v.z);
  atomicAdd(p + 3, v.w);
  if (lane == 0) atomicAdd(cnt + g, 1.0f);
}

__global__ void pool_div_kernel(float* pool, const float* cnt, int total) {
  int i = blockIdx.x * blockDim.x + threadIdx.x;
  if (i < total) pool[i] /= fmaxf(cnt[i >> 7], 1.0f);
}

// ---------------- small dense MLP layers ------------------------------------
__global__ void mlp_kernel(const float* __restrict__ in, const float* __restrict__ w,
                           const float* __restrict__ b, float* __restrict__ out,
                           int rows, int ncols, int do_relu) {
  int gid = blockIdx.x * blockDim.x + threadIdx.x;
  int r = gid / ncols, j = gid - r * ncols;
  if (r >= rows) return;
  float s = b[j];
  const float* ip = in + (size_t)r * F;
  const float* wp = w + (size_t)j * F;
#pragma unroll 4
  for (int k = 0; k < F; ++k) s += ip[k] * wp[k];
  out[(size_t)r * ncols + j] = do_relu ? fmaxf(s, 0.0f) : s;
}

extern "C" void kernel_launch(void* const* d_in, const int* in_sizes, int n_in,
                              void* d_out, int out_size, void* d_ws, size_t ws_size,
                              hipStream_t stream) {
  const float* x      = (const float*)d_in[0];
  const int*   ei     = (const int*)d_in[1];
  const float* ew     = (const float*)d_in[2];
  const int*   batch  = (const int*)d_in[3];
  const float* W1     = (const float*)d_in[4];
  const float* b1     = (const float*)d_in[5];
  const float* W2     = (const float*)d_in[6];
  const float* b2     = (const float*)d_in[7];
  const float* lin1_w = (const float*)d_in[8];
  const float* lin1_b = (const float*)d_in[9];
  const float* lin2_w = (const float*)d_in[10];
  const float* lin2_b = (const float*)d_in[11];
  float* out = (float*)d_out;

  const int N = in_sizes[0] / F;     // 50000
  const int E = in_sizes[1] / 2;     // 800000
  const int G = out_size / 2;        // 100 graphs
  const int* row = ei;
  const int* col = ei + E;

  // workspace layout (floats)
  float* ws   = (float*)d_ws;
  float* dinv = ws;                                   // N
  float* bufA = dinv + (((size_t)N + 63) & ~(size_t)63);
  float* bufB = bufA + (size_t)N * F;
  float* pool = bufB + (size_t)N * F;                 // G*F
  float* cnt  = pool + (size_t)G * F;                 // G   (contiguous w/ pool)
  float* g1   = cnt + (((size_t)G + 63) & ~(size_t)63);

  const int T = 256;
  dim3 bt(T);

  // normalization coefficients
  deg_init_kernel<<<(N + T - 1) / T, bt, 0, stream>>>(dinv, N);
  deg_edge_kernel<<<(E + T - 1) / T, bt, 0, stream>>>(dinv, col, ew, E);
  dinv_kernel<<<(N + T - 1) / T, bt, 0, stream>>>(dinv, N);

  int gemm_blocks = (N + 15) / 16;
  long long nv = (long long)N * 32;  // one wave per node row
  long long ev = (long long)E * 32;  // one wave per edge

  // layer 1: h = relu(agg(x @ W1^T) + b1)
  gemm_wmma_f32<<<gemm_blocks, 32, 0, stream>>>(x, W1, bufA, N);
  selfloop_init_kernel<<<(nv + T - 1) / T, bt, 0, stream>>>(bufA, dinv, bufB, N);
  agg_edge_kernel<<<(ev + T - 1) / T, bt, 0, stream>>>(bufA, row, col, ew, dinv, bufB, E);
  bias_relu_kernel<<<(nv + T - 1) / T, bt, 0, stream>>>(bufB, b1, N);

  // layer 2: h = relu(agg(h @ W2^T) + b2)
  gemm_wmma_f32<<<gemm_blocks, 32, 0, stream>>>(bufB, W2, bufA, N);
  selfloop_init_kernel<<<(nv + T - 1) / T, bt, 0, stream>>>(bufA, dinv, bufB, N);
  agg_edge_kernel<<<(ev + T - 1) / T, bt, 0, stream>>>(bufA, row, col, ew, dinv, bufB, E);
  bias_relu_kernel<<<(nv + T - 1) / T, bt, 0, stream>>>(bufB, b2, N);

  // global mean pool
  int pz = G * F + G;                // pool + cnt contiguous
  zero_kernel<<<(pz + T - 1) / T, bt, 0, stream>>>(pool, pz);
  pool_add_kernel<<<(nv + T - 1) / T, bt, 0, stream>>>(bufB, batch, pool, cnt, N);
  pool_div_kernel<<<(G * F + T - 1) / T, bt, 0, stream>>>(pool, cnt, G * F);

  // head MLP
  mlp_kernel<<<(G * F + T - 1) / T, bt, 0, stream>>>(pool, lin1_w, lin1_b, g1, G, F, 1);
  mlp_kernel<<<(G * 2 + T - 1) / T, bt, 0, stream>>>(g1, lin2_w, lin2_b, out, G, 2, 0);
}